// RIIDModelLSTM_11708080848981
// MI455X (gfx1250) — compile-verified
//
#include <hip/hip_runtime.h>
#include <cstddef>

// Problem constants (match reference)
#define LSTM_H   6
#define LSTM_T   512
#define LSTM_B   8192
#define LSTM_NG  24   // 4*H gates

typedef float v2f __attribute__((ext_vector_type(2)));
typedef float v8f __attribute__((ext_vector_type(8)));

// Fast activations on the hardware transcendental pipe (1-ulp class).
__device__ __forceinline__ float fexp2(float x) { return __builtin_amdgcn_exp2f(x); }
__device__ __forceinline__ float frcp(float x)  { return __builtin_amdgcn_rcpf(x); }
__device__ __forceinline__ float sigf(float x) {
  return frcp(1.0f + fexp2(-1.44269504089f * x));               // 1/(1+e^-x)
}
__device__ __forceinline__ float tanhfast(float x) {
  return 1.0f - 2.0f * frcp(1.0f + fexp2(2.88539008178f * x));  // 1 - 2/(1+e^{2x})
}
__device__ __forceinline__ float relu(float x) { return x > 0.0f ? x : 0.0f; }

// ds_swizzle xor pattern, group-of-32 mode: offset[15]=0, xor_mask[14:10], and_mask=0x1F
#define SWZ_XOR(m) ((((m) & 31) << 10) | 0x1F)

template <int PAT>
__device__ __forceinline__ float swzf(float v) {
  return __int_as_float(__builtin_amdgcn_ds_swizzle(__float_as_int(v), PAT));
}

// Joint reduction step: sums 'a' on lanes with (tid&MASK)==0 and 'b' on the others,
// each over the lane pair joined by xor-MASK.
template <int MASK>
__device__ __forceinline__ float combine2(float a, float b, int tid) {
  const bool up = (tid & MASK) != 0;
  const float t = up ? a : b;                 // value handed to partner
  const float u = swzf<SWZ_XOR(MASK)>(t);     // partner's handed value
  return (up ? b : a) + u;
}

__global__ __launch_bounds__(32) void lstm2_fc_fused(
    const float* __restrict__ x,
    const float* __restrict__ w_ih_0, const float* __restrict__ w_hh_0,
    const float* __restrict__ b_ih_0, const float* __restrict__ b_hh_0,
    const float* __restrict__ w_ih_1, const float* __restrict__ w_hh_1,
    const float* __restrict__ b_ih_1, const float* __restrict__ b_hh_1,
    const float* __restrict__ fc1_w,  const float* __restrict__ fc1_b,
    const float* __restrict__ fc2_w,  const float* __restrict__ fc2_b,
    float* __restrict__ out)
{
  constexpr int H  = LSTM_H;
  constexpr int T  = LSTM_T;
  constexpr int NG = LSTM_NG;

  // LDS: packed LSTM weights [gate][12] = [w_ih row | w_hh row] + combined biases
  __shared__ __align__(16) float lw0[NG * 12];
  __shared__ __align__(16) float lw1[NG * 12];
  __shared__ float lb0[NG];
  __shared__ float lb1[NG];

  const int tid   = threadIdx.x;   // 0..31
  const int lm    = tid & 15;      // sequence slot within wave / WMMA M-row / N-lane
  const int hi    = tid >> 4;      // lane-pair half: owns hidden units 3*hi..3*hi+2
  const int gbase = blockIdx.x * 16;           // 16 sequences per wave
  const int seq   = gbase + lm;

  for (int i = tid; i < NG * 12; i += 32) {
    const int g = i / 12, k = i % 12;
    lw0[i] = (k < H) ? w_ih_0[g * H + k] : w_hh_0[g * H + (k - H)];
    lw1[i] = (k < H) ? w_ih_1[g * H + k] : w_hh_1[g * H + (k - H)];
  }
  if (tid < NG) {
    lb0[tid] = b_ih_0[tid] + b_hh_0[tid];
    lb1[tid] = b_ih_1[tid] + b_hh_1[tid];
  }
  __syncthreads();

  // ---- FC1 weights resident in VGPRs in f32 WMMA B-matrix layout ----
  // B (4x16 f32): VGPR0 lane L = B[2*(L/16)][L%16], VGPR1 = next K.
  v2f Bk[2][2];  // [n-tile][k-chunk]
#pragma unroll
  for (int nt = 0; nt < 2; ++nt) {
#pragma unroll
    for (int kc = 0; kc < 2; ++kc) {
      const int n  = nt * 16 + lm;
      const int k0 = kc * 4 + 2 * hi;
      v2f b;
      b.x = (k0     < H) ? fc1_w[n * H + k0    ] : 0.0f;
      b.y = (k0 + 1 < H) ? fc1_w[n * H + k0 + 1] : 0.0f;
      Bk[nt][kc] = b;
    }
  }
  const float bn0 = fc1_b[lm];
  const float bn1 = fc1_b[lm + 16];
  const float w2a = fc2_w[lm];
  const float w2b = fc2_w[lm + 16];
  const float b2  = fc2_b[0];

  // ---- preload this lane's 12 gate rows per layer into registers ----
  // local gate index li = q*3 + j  <->  global gate g = q*6 + (3*hi + j), q in {i,f,g,o}
  const int kbase = 3 * hi;
  float wr0[12][12], wr1[12][12], br0[12], br1[12];
#pragma unroll
  for (int j = 0; j < 3; ++j) {
#pragma unroll
    for (int q = 0; q < 4; ++q) {
      const int g  = q * 6 + kbase + j;
      const int li = q * 3 + j;
      br0[li] = lb0[g];
      br1[li] = lb1[g];
#pragma unroll
      for (int k = 0; k < 12; ++k) {
        wr0[li][k] = lw0[g * 12 + k];
        wr1[li][k] = lw1[g * 12 + k];
      }
    }
  }

  // ---- LSTM state ----
  float h0f[H], h1f[H];   // full hidden vectors (identical across the lane pair)
  float c0o[3], c1o[3];   // own cell states (3 units per lane)
#pragma unroll
  for (int k = 0; k < H; ++k) { h0f[k] = 0.f; h1f[k] = 0.f; }
#pragma unroll
  for (int j = 0; j < 3; ++j) { c0o[j] = 0.f; c1o[j] = 0.f; }

  const float* xrow = x + (size_t)seq * T * H;

#pragma unroll 1
  for (int t = 0; t < T; ++t) {
    // x_t (both lanes of a pair load the same row; cache hit for the partner)
    float xv[H];
    {
      const v2f* xp = (const v2f*)(xrow + (size_t)t * H);
      const v2f a = xp[0], b = xp[1], c = xp[2];
      xv[0] = a.x; xv[1] = a.y; xv[2] = b.x; xv[3] = b.y; xv[4] = c.x; xv[5] = c.y;
    }
    __builtin_prefetch(xrow + (size_t)(t + 1) * H, 0, 1);  // global_prefetch next row

    float acc[4][3], own[3];

    // ---- layer 0: this lane's 12 gates ----
#pragma unroll
    for (int j = 0; j < 3; ++j) {
#pragma unroll
      for (int q = 0; q < 4; ++q) {
        const int li = q * 3 + j;
        float a = br0[li];
#pragma unroll
        for (int k = 0; k < H; ++k) a += xv[k] * wr0[li][k];
#pragma unroll
        for (int k = 0; k < H; ++k) a += h0f[k] * wr0[li][H + k];
        acc[q][j] = a;
      }
    }
#pragma unroll
    for (int j = 0; j < 3; ++j) {
      const float ig = sigf(acc[0][j]);
      const float fg = sigf(acc[1][j]);
      const float gg = tanhfast(acc[2][j]);
      const float og = sigf(acc[3][j]);
      c0o[j] = fg * c0o[j] + ig * gg;
      own[j] = og * tanhfast(c0o[j]);
    }
#pragma unroll
    for (int j = 0; j < 3; ++j) {   // exchange h0 halves across the lane pair (xor 16)
      const float oth = swzf<SWZ_XOR(16)>(own[j]);
      h0f[j]     = hi ? oth    : own[j];
      h0f[3 + j] = hi ? own[j] : oth;
    }

    // ---- layer 1 ----
#pragma unroll
    for (int j = 0; j < 3; ++j) {
#pragma unroll
      for (int q = 0; q < 4; ++q) {
        const int li = q * 3 + j;
        float a = br1[li];
#pragma unroll
        for (int k = 0; k < H; ++k) a += h0f[k] * wr1[li][k];
#pragma unroll
        for (int k = 0; k < H; ++k) a += h1f[k] * wr1[li][H + k];
        acc[q][j] = a;
      }
    }
#pragma unroll
    for (int j = 0; j < 3; ++j) {
      const float ig = sigf(acc[0][j]);
      const float fg = sigf(acc[1][j]);
      const float gg = tanhfast(acc[2][j]);
      const float og = sigf(acc[3][j]);
      c1o[j] = fg * c1o[j] + ig * gg;
      own[j] = og * tanhfast(c1o[j]);
    }
#pragma unroll
    for (int j = 0; j < 3; ++j) {
      const float oth = swzf<SWZ_XOR(16)>(own[j]);
      h1f[j]     = hi ? oth    : own[j];
      h1f[3 + j] = hi ? own[j] : oth;
    }

    // ---- FC1 via fp32 WMMA: [16 x 6] @ [6 x 32], K padded to 8 ----
    // A-layout: lane L holds row M = L%16 (this lane's own sequence!), K = 2*(L/16)+{0,1}.
    // Every lane already has the full h1f vector, so A is built with 4 selects — no LDS.
    v2f a0, a1;
    a0.x = hi ? h1f[2] : h1f[0];
    a0.y = hi ? h1f[3] : h1f[1];
    a1.x = hi ? 0.0f   : h1f[4];   // K blocks 4..7: slots 6,7 are zero pad
    a1.y = hi ? 0.0f   : h1f[5];

    v8f cA, cB;
#pragma unroll
    for (int r = 0; r < 8; ++r) { cA[r] = bn0; cB[r] = bn1; }
    cA = __builtin_amdgcn_wmma_f32_16x16x4_f32(false, a0, false, Bk[0][0], (short)0, cA, false, false);
    cA = __builtin_amdgcn_wmma_f32_16x16x4_f32(false, a1, false, Bk[0][1], (short)0, cA, false, false);
    cB = __builtin_amdgcn_wmma_f32_16x16x4_f32(false, a0, false, Bk[1][0], (short)0, cB, false, false);
    cB = __builtin_amdgcn_wmma_f32_16x16x4_f32(false, a1, false, Bk[1][1], (short)0, cB, false, false);

    // ---- FC2 + relu, joint tree reduction ----
    // v[r] at lane: row m = r + 8*hi, n = lm (tile A) and n = 16+lm (tile B)
    float v0 = relu(cA[0]) * w2a + relu(cB[0]) * w2b;
    float v1 = relu(cA[1]) * w2a + relu(cB[1]) * w2b;
    float v2 = relu(cA[2]) * w2a + relu(cB[2]) * w2b;
    float v3 = relu(cA[3]) * w2a + relu(cB[3]) * w2b;
    float v4 = relu(cA[4]) * w2a + relu(cB[4]) * w2b;
    float v5 = relu(cA[5]) * w2a + relu(cB[5]) * w2b;
    float v6 = relu(cA[6]) * w2a + relu(cB[6]) * w2b;
    float v7 = relu(cA[7]) * w2a + relu(cB[7]) * w2b;

    const float p0 = combine2<1>(v0, v1, tid);
    const float p1 = combine2<1>(v2, v3, tid);
    const float p2 = combine2<1>(v4, v5, tid);
    const float p3 = combine2<1>(v6, v7, tid);
    const float q0 = combine2<2>(p0, p1, tid);
    const float q1 = combine2<2>(p2, p3, tid);
    float z = combine2<4>(q0, q1, tid);
    z += swzf<SWZ_XOR(8)>(z);   // finish the 16-lane sum within each half

    // lane j holds the full sum for row m = (j&7) + 8*hi; one coalesced store
    const float ov = relu(z + b2);
    if ((tid & 8) == 0) {
      const int m = (tid & 7) + 8 * hi;
      out[(size_t)(gbase + m) * T + t] = ov;
    }
  }
}

extern "C" void kernel_launch(void* const* d_in, const int* in_sizes, int n_in,
                              void* d_out, int out_size, void* d_ws, size_t ws_size,
                              hipStream_t stream) {
  (void)in_sizes; (void)n_in; (void)d_ws; (void)ws_size; (void)out_size;
  const float* x      = (const float*)d_in[0];
  const float* w_ih_0 = (const float*)d_in[1];
  const float* w_hh_0 = (const float*)d_in[2];
  const float* b_ih_0 = (const float*)d_in[3];
  const float* b_hh_0 = (const float*)d_in[4];
  const float* w_ih_1 = (const float*)d_in[5];
  const float* w_hh_1 = (const float*)d_in[6];
  const float* b_ih_1 = (const float*)d_in[7];
  const float* b_hh_1 = (const float*)d_in[8];
  const float* fc1_w  = (const float*)d_in[9];
  const float* fc1_b  = (const float*)d_in[10];
  const float* fc2_w  = (const float*)d_in[11];
  const float* fc2_b  = (const float*)d_in[12];
  float* out = (float*)d_out;

  dim3 grid(LSTM_B / 16);   // 512 single-wave workgroups, 16 sequences each
  dim3 block(32);
  lstm2_fc_fused<<<grid, block, 0, stream>>>(
      x, w_ih_0, w_hh_0, b_ih_0, b_hh_0,
      w_ih_1, w_hh_1, b_ih_1, b_hh_1,
      fc1_w, fc1_b, fc2_w, fc2_b, out);
}